// GCN_74818330296984
// MI455X (gfx1250) — compile-verified
//
#include <hip/hip_runtime.h>

typedef __attribute__((ext_vector_type(2))) float v2f;
typedef __attribute__((ext_vector_type(8))) float v8f;

#define N_NODES   50000
#define N_EDGES   800000
#define IN_FEATS  256
#define H_FEATS   128
#define N_CLASSES 16

// ---------------------------------------------------------------- utilities
__global__ void gcn_zero_kernel(float* __restrict__ p, int n) {
    int i = blockIdx.x * blockDim.x + threadIdx.x;
    if (i < n) p[i] = 0.0f;
}

__global__ void gcn_degree_kernel(const int* __restrict__ src,
                                  const int* __restrict__ dst,
                                  float* __restrict__ deg_out,
                                  float* __restrict__ deg_in) {
    int e = blockIdx.x * blockDim.x + threadIdx.x;
    if (e < N_EDGES) {
        atomicAdd(&deg_out[src[e]], 1.0f);
        atomicAdd(&deg_in[dst[e]], 1.0f);
    }
}

__global__ void gcn_norm_kernel(const float* __restrict__ deg,
                                float* __restrict__ nrm, int n) {
    int i = blockIdx.x * blockDim.x + threadIdx.x;
    if (i < n) {
        float d = deg[i];
        nrm[i] = (d > 0.0f) ? rsqrtf(fmaxf(d, 1.0f)) : 0.0f;
    }
}

// ---------------------------------------------------------------- GEMM 1
// xw[r, 0:128] = (feat[r, 0:256] * norm_out[r]) @ W1[256,128]
// 8 waves / block, one 16-row strip per wave. W1 staged in LDS as 2x64KB
// K-tiles, pre-interleaved by row pairs so each lane's B fragment
// (W[kk][c], W[kk+1][c]) is one contiguous 64-bit ds load.
__global__ __launch_bounds__(256) void gcn_gemm1_kernel(
        const float* __restrict__ feat, const float* __restrict__ W1,
        const float* __restrict__ norm_out, float* __restrict__ xw) {
    __shared__ float sW[128 * H_FEATS];        // 64 KB K-tile (interleaved)

    const int wave = threadIdx.x >> 5;
    const int lane = threadIdx.x & 31;
    const int half = lane >> 4;                // 0: K+0..1, 1: K+2..3
    const int l16  = lane & 15;

    const int row_base = blockIdx.x * 128 + wave * 16;
    const int row      = row_base + l16;
    const float nrm    = (row < N_NODES) ? norm_out[row] : 0.0f;
    const float* frow  = feat + (size_t)((row < N_NODES) ? row : 0) * IN_FEATS;

    v8f acc[8] = {};

    for (int kb = 0; kb < IN_FEATS; kb += 128) {
        __syncthreads();
        // stage rows [kb, kb+128) of W1, interleaved by row pairs:
        // sW[(k2*H_FEATS + c)*2 + parity] = W1[kb + 2*k2 + parity][c]
        for (int t = threadIdx.x; t < 64 * (H_FEATS / 4); t += 256) {
            int k2 = t >> 5;                 // row pair 0..63
            int c4 = (t & 31) * 4;           // col group
            const float* r0 = W1 + (size_t)(kb + 2 * k2) * H_FEATS + c4;
            float4 a0 = *(const float4*)r0;
            float4 a1 = *(const float4*)(r0 + H_FEATS);
            float2* o = (float2*)&sW[(k2 * H_FEATS + c4) * 2];
            o[0] = make_float2(a0.x, a1.x);
            o[1] = make_float2(a0.y, a1.y);
            o[2] = make_float2(a0.z, a1.z);
            o[3] = make_float2(a0.w, a1.w);
        }
        // prefetch next A-stream K-chunk while this tile computes
        if (kb + 128 < IN_FEATS) __builtin_prefetch(frow + kb + 128, 0, 1);
        __syncthreads();

        for (int k = 0; k < 128; k += 4) {
            const int kk = k + 2 * half;
            const int kp = (k >> 1) + half;  // row-pair index of (kk, kk+1)
            v2f a;
            a.x = frow[kb + kk]     * nrm;
            a.y = frow[kb + kk + 1] * nrm;
            const v2f* brow = (const v2f*)&sW[kp * (H_FEATS * 2)];
#pragma unroll
            for (int n = 0; n < 8; ++n) {
                v2f b = brow[n * 16 + l16];  // single ds_load_b64
                acc[n] = __builtin_amdgcn_wmma_f32_16x16x4_f32(
                    false, a, false, b, (short)0, acc[n], false, false);
            }
        }
    }

#pragma unroll
    for (int n = 0; n < 8; ++n) {
#pragma unroll
        for (int i = 0; i < 8; ++i) {
            int r = row_base + i + 8 * half; // C layout: VGPR i -> M=i (+8 upper half)
            if (r < N_NODES) xw[(size_t)r * H_FEATS + n * 16 + l16] = acc[n][i];
        }
    }
}

// ---------------------------------------------------------------- GEMM 2
// xw2[r, 0:16] = (h[r, 0:128] * norm_out[r]) @ W2[128,16]
__global__ __launch_bounds__(256) void gcn_gemm2_kernel(
        const float* __restrict__ h, const float* __restrict__ W2,
        const float* __restrict__ norm_out, float* __restrict__ xw2) {
    __shared__ float sW[H_FEATS * N_CLASSES];  // 8 KB (interleaved row pairs)

    {
        int t = threadIdx.x;                   // exactly 256 chunks
        int k2 = t >> 2;                       // row pair 0..63
        int c4 = (t & 3) * 4;                  // col group
        const float* r0 = W2 + (size_t)(2 * k2) * N_CLASSES + c4;
        float4 a0 = *(const float4*)r0;
        float4 a1 = *(const float4*)(r0 + N_CLASSES);
        float2* o = (float2*)&sW[(k2 * N_CLASSES + c4) * 2];
        o[0] = make_float2(a0.x, a1.x);
        o[1] = make_float2(a0.y, a1.y);
        o[2] = make_float2(a0.z, a1.z);
        o[3] = make_float2(a0.w, a1.w);
    }
    __syncthreads();

    const int wave = threadIdx.x >> 5;
    const int lane = threadIdx.x & 31;
    const int half = lane >> 4;
    const int l16  = lane & 15;

    const int row_base = blockIdx.x * 128 + wave * 16;
    const int row      = row_base + l16;
    const float nrm    = (row < N_NODES) ? norm_out[row] : 0.0f;
    const float* hrow  = h + (size_t)((row < N_NODES) ? row : 0) * H_FEATS;

    v8f acc = {};
    for (int k = 0; k < H_FEATS; k += 4) {
        const int kk = k + 2 * half;
        const int kp = (k >> 1) + half;
        v2f a;
        a.x = hrow[kk]     * nrm;
        a.y = hrow[kk + 1] * nrm;
        v2f b = ((const v2f*)&sW[kp * (N_CLASSES * 2)])[l16];
        acc = __builtin_amdgcn_wmma_f32_16x16x4_f32(
            false, a, false, b, (short)0, acc, false, false);
    }

#pragma unroll
    for (int i = 0; i < 8; ++i) {
        int r = row_base + i + 8 * half;
        if (r < N_NODES) xw2[(size_t)r * N_CLASSES + l16] = acc[i];
    }
}

// ---------------------------------------------------------------- scatter
// agg[dst[e], :] += xw[src[e], :]  (feature-vectorized: float4 load + 4 atomics)
__global__ void gcn_scatter1_kernel(const int* __restrict__ src,
                                    const int* __restrict__ dst,
                                    const float* __restrict__ xw,
                                    float* __restrict__ agg) {
    int idx = blockIdx.x * blockDim.x + threadIdx.x;      // N_EDGES * 32
    if (idx >= N_EDGES * 32) return;
    int e = idx >> 5, c = idx & 31;
    int s = src[e], d = dst[e];
    float4 v = *(const float4*)(xw + (size_t)s * H_FEATS + c * 4);
    float* o = agg + (size_t)d * H_FEATS + c * 4;
    atomicAdd(o + 0, v.x); atomicAdd(o + 1, v.y);
    atomicAdd(o + 2, v.z); atomicAdd(o + 3, v.w);
}

__global__ void gcn_scatter2_kernel(const int* __restrict__ src,
                                    const int* __restrict__ dst,
                                    const float* __restrict__ xw2,
                                    float* __restrict__ out) {
    int idx = blockIdx.x * blockDim.x + threadIdx.x;      // N_EDGES * 4
    if (idx >= N_EDGES * 4) return;
    int e = idx >> 2, c = idx & 3;
    int s = src[e], d = dst[e];
    float4 v = *(const float4*)(xw2 + (size_t)s * N_CLASSES + c * 4);
    float* o = out + (size_t)d * N_CLASSES + c * 4;
    atomicAdd(o + 0, v.x); atomicAdd(o + 1, v.y);
    atomicAdd(o + 2, v.z); atomicAdd(o + 3, v.w);
}

// h = relu(agg * norm_in + b1)
__global__ void gcn_finish1_kernel(const float* __restrict__ agg,
                                   const float* __restrict__ norm_in,
                                   const float* __restrict__ b1,
                                   float* __restrict__ hout) {
    int idx = blockIdx.x * blockDim.x + threadIdx.x;      // N_NODES * 128
    if (idx >= N_NODES * H_FEATS) return;
    int r = idx >> 7, f = idx & (H_FEATS - 1);
    float v = agg[idx] * norm_in[r] + b1[f];
    hout[idx] = fmaxf(v, 0.0f);
}

// out = log_softmax(out * norm_in + b2) per node (in place)
__global__ void gcn_final_kernel(float* __restrict__ out,
                                 const float* __restrict__ norm_in,
                                 const float* __restrict__ b2) {
    int nid = blockIdx.x * blockDim.x + threadIdx.x;
    if (nid >= N_NODES) return;
    float ni = norm_in[nid];
    float v[N_CLASSES];
    float m = -3.402823466e+38f;
#pragma unroll
    for (int f = 0; f < N_CLASSES; ++f) {
        v[f] = out[(size_t)nid * N_CLASSES + f] * ni + b2[f];
        m = fmaxf(m, v[f]);
    }
    float s = 0.0f;
#pragma unroll
    for (int f = 0; f < N_CLASSES; ++f) s += expf(v[f] - m);
    float lse = logf(s);
#pragma unroll
    for (int f = 0; f < N_CLASSES; ++f)
        out[(size_t)nid * N_CLASSES + f] = v[f] - m - lse;
}

// ---------------------------------------------------------------- launch
extern "C" void kernel_launch(void* const* d_in, const int* in_sizes, int n_in,
                              void* d_out, int out_size, void* d_ws, size_t ws_size,
                              hipStream_t stream) {
    const float* feat = (const float*)d_in[0];
    const int*   src  = (const int*)d_in[1];
    const int*   dst  = (const int*)d_in[2];
    const float* W1   = (const float*)d_in[3];
    const float* b1   = (const float*)d_in[4];
    const float* W2   = (const float*)d_in[5];
    const float* b2   = (const float*)d_in[6];
    float* out = (float*)d_out;
    float* ws  = (float*)d_ws;

    // ws layout (floats). [0 .. 6.5M) must be zeroed each call.
    float* deg_out  = ws;                                 //   50000
    float* deg_in   = deg_out + N_NODES;                  //   50000
    float* agg1     = deg_in + N_NODES;                   // 6400000
    float* norm_out = agg1 + (size_t)N_NODES * H_FEATS;   //   50000
    float* norm_in  = norm_out + N_NODES;                 //   50000
    float* bufA     = norm_in + N_NODES;                  // xw1, then reused for h
    float* xw2      = bufA + (size_t)N_NODES * H_FEATS;   //  800000

    const int ZEROED = 2 * N_NODES + N_NODES * H_FEATS;   // deg_out, deg_in, agg1

    gcn_zero_kernel<<<(ZEROED + 255) / 256, 256, 0, stream>>>(ws, ZEROED);
    gcn_zero_kernel<<<(N_NODES * N_CLASSES + 255) / 256, 256, 0, stream>>>(out, N_NODES * N_CLASSES);

    gcn_degree_kernel<<<(N_EDGES + 255) / 256, 256, 0, stream>>>(src, dst, deg_out, deg_in);
    gcn_norm_kernel<<<(N_NODES + 255) / 256, 256, 0, stream>>>(deg_out, norm_out, N_NODES);
    gcn_norm_kernel<<<(N_NODES + 255) / 256, 256, 0, stream>>>(deg_in, norm_in, N_NODES);

    const int row_blocks = (N_NODES + 127) / 128;  // 391
    gcn_gemm1_kernel<<<row_blocks, 256, 0, stream>>>(feat, W1, norm_out, bufA);
    gcn_scatter1_kernel<<<(N_EDGES * 32 + 255) / 256, 256, 0, stream>>>(src, dst, bufA, agg1);
    gcn_finish1_kernel<<<(N_NODES * H_FEATS + 255) / 256, 256, 0, stream>>>(agg1, norm_in, b1, bufA);

    gcn_gemm2_kernel<<<row_blocks, 256, 0, stream>>>(bufA, W2, norm_out, xw2);
    gcn_scatter2_kernel<<<(N_EDGES * 4 + 255) / 256, 256, 0, stream>>>(src, dst, xw2, out);
    gcn_final_kernel<<<(N_NODES + 255) / 256, 256, 0, stream>>>(out, norm_in, b2);
}